// AttentionSubModule_42339787604121
// MI455X (gfx1250) — compile-verified
//
#include <hip/hip_runtime.h>
#include <cstddef>

typedef __attribute__((ext_vector_type(2))) float v2f;
typedef __attribute__((ext_vector_type(8))) float v8f;

#define WPB 4                  // waves (= batch elements) per block
#define NTHREADS (WPB * 32)
#define SET_STRIDE 468         // w_j(81)+b_j(9)+w_o(153)+b_o(9)+w_g(99)+b_g(9)+w_b(99)+b_b(9)
#define W_TOTAL (3 * SET_STRIDE)
#define LN_EPS 1e-5f

struct Ptrs { const float* p[27]; };

// D = A(16x4) * B(4x16) + C(16x16), full fp32 (V_WMMA_F32_16X16X4_F32)
#define WMMA4(A, B, C) \
    __builtin_amdgcn_wmma_f32_16x16x4_f32(false, (A), false, (B), (short)0, (C), false, false)

// ---- DPP16 helpers: reductions confined to 16-lane rows (VALU pipe, no LDS) ----
template <int CTRL>
__device__ __forceinline__ float dppf(float v) {
    return __int_as_float(__builtin_amdgcn_update_dpp(
        0, __float_as_int(v), CTRL, 0xF, 0xF, true));
}
// butterfly/rotate reduce: after quad_perm(1,0,3,2), quad_perm(2,3,0,1),
// row_ror:4, row_ror:8 every lane of a 16-lane row holds the row-wide result.
__device__ __forceinline__ float rowsum16(float s) {
    s += dppf<0xB1>(s);      // quad_perm [1,0,3,2]
    s += dppf<0x4E>(s);      // quad_perm [2,3,0,1]
    s += dppf<0x124>(s);     // row_ror:4
    s += dppf<0x128>(s);     // row_ror:8
    return s;
}
__device__ __forceinline__ float rowmax16(float s) {
    s = fmaxf(s, dppf<0xB1>(s));
    s = fmaxf(s, dppf<0x4E>(s));
    s = fmaxf(s, dppf<0x124>(s));
    s = fmaxf(s, dppf<0x128>(s));
    return s;
}

__global__ __launch_bounds__(NTHREADS)
void attn_sub_module_kernel(Ptrs P, float* __restrict__ out, int nB)
{
    // Per-block LDS: shared weights + per-wave staging (padded, zero-filled pads)
    __shared__ __align__(16) float sW[W_TOTAL];
    __shared__ __align__(16) float sKeys[WPB][32 * 12]; // rows 0..31 (25 real), cols 0..11 (9 real)
    __shared__ __align__(16) float sRes [WPB][32 * 16]; // C-init for second WMMA
    __shared__ __align__(16) float sValT[WPB][9 * 32];  // V transposed: [kv][slot]
    __shared__ __align__(16) float sAttn[WPB][32 * 32]; // softmax result, A-matrix of 2nd WMMA

    // ---- stage all 24 weight/bias arrays into LDS (d_in[1..24] order == LDS order) ----
    {
        const int sz8[8] = {81, 9, 153, 9, 99, 9, 99, 9};
        int off = 0;
        for (int a = 0; a < 24; ++a) {
            int sz = sz8[a & 7];
            const float* src = P.p[1 + a];
            for (int i = threadIdx.x; i < sz; i += NTHREADS) sW[off + i] = src[i];
            off += sz;
        }
    }
    __syncthreads();

    const int lane = threadIdx.x & 31;
    const int wv   = threadIdx.x >> 5;
    const int hi   = lane >> 4;        // which 16-lane half
    const int l15  = lane & 15;
    const bool k9  = (l15 < 9);

    long long b = (long long)blockIdx.x * WPB + wv;
    if (b >= nB) return;               // uniform per wave

    const float* x = P.p[0];

    // ================= projections: lane q owns slot row q (q < 25) =================
    {
        int q = lane;
        int seg = (q < 3) ? 0 : (q < 13) ? 1 : (q < 23) ? 2 : 3;
        const int dins[4] = {9, 17, 11, 11};
        const int offw[4] = {0, 90, 252, 360};
        int din  = dins[seg];
        int xoff = (seg == 0) ? q * 9
                 : (seg == 1) ? 27 + (q - 3) * 17
                 : (seg == 2) ? 197 + (q - 13) * 11
                 :              307 + (q - 23) * 11;

        float xr[17];
        float ko[9], vo[9], ro[9];
        #pragma unroll
        for (int k = 0; k < 9; ++k) { ko[k] = 0.f; vo[k] = 0.f; ro[k] = 0.f; }

        if (q < 25) {
            const float* xp = x + (size_t)b * 329 + xoff;
            #pragma unroll
            for (int d = 0; d < 17; ++d) {        // zero-padded segment in registers
                float t = xp[(d < din) ? d : 0];
                xr[d] = (d < din) ? t : 0.f;
            }
            #pragma unroll
            for (int s = 0; s < 3; ++s) {         // K, V, R sets
                const int wb = s * SET_STRIDE + offw[seg];
                const int bb = wb + 9 * din;
                #pragma unroll
                for (int k = 0; k < 9; ++k) {
                    float acc = sW[bb + k];
                    const int r0 = wb + k * din;
                    #pragma unroll
                    for (int d = 0; d < 17; ++d) acc += xr[d] * sW[r0 + d]; // pads: xr==0
                    if (s == 0) ko[k] = acc; else if (s == 1) vo[k] = acc; else ro[k] = acc;
                }
            }
        }
        // all 32 lanes store (lanes q>=25 store zeros -> pad rows/cols are exactly 0)
        float4* kr = (float4*)&sKeys[wv][q * 12];
        kr[0] = make_float4(ko[0], ko[1], ko[2], ko[3]);
        kr[1] = make_float4(ko[4], ko[5], ko[6], ko[7]);
        kr[2] = make_float4(ko[8], 0.f, 0.f, 0.f);
        float4* rr = (float4*)&sRes[wv][q * 16];
        rr[0] = make_float4(ro[0], ro[1], ro[2], ro[3]);
        rr[1] = make_float4(ro[4], ro[5], ro[6], ro[7]);
        rr[2] = make_float4(ro[8], 0.f, 0.f, 0.f);
        rr[3] = make_float4(0.f, 0.f, 0.f, 0.f);
        #pragma unroll
        for (int k = 0; k < 9; ++k) sValT[wv][k * 32 + q] = vo[k];  // cols 25..31 zeroed
    }
    asm volatile("" ::: "memory"); // same-wave LDS ops are in-order; block compiler reordering

    // ================= scores = K * K^T  (M=N=32 padded, K=12 padded), fp32 WMMA =====
    // A-tile(mt) and B-tile(nt) of a symmetric product read identical LDS words.
    const float* Kp = &sKeys[wv][0];
    v8f s00 = {0,0,0,0,0,0,0,0};
    v8f s01 = s00, s10 = s00, s11 = s00;
    #pragma unroll
    for (int kc = 0; kc < 3; ++kc) {
        int kk = kc * 4 + hi * 2;
        v2f a0 = *(const v2f*)&Kp[(l15)      * 12 + kk];
        v2f a1 = *(const v2f*)&Kp[(16 + l15) * 12 + kk];
        s00 = WMMA4(a0, a0, s00);
        s01 = WMMA4(a0, a1, s01);
        s10 = WMMA4(a1, a0, s10);
        s11 = WMMA4(a1, a1, s11);
    }

    // ================= softmax over 25 cols (scale 1/sqrt(9) folded into exp arg) ====
    const float inv3 = 1.0f / 3.0f;
    #pragma unroll
    for (int mt = 0; mt < 2; ++mt) {
        v8f e0 = mt ? s10 : s00;   // cols 0..15
        v8f e1 = mt ? s11 : s01;   // cols 16..31 (valid: 16..24)
        #pragma unroll
        for (int r = 0; r < 8; ++r) {
            float va = e0[r];
            float vb = k9 ? e1[r] : -3.0e38f;
            float m  = rowmax16(fmaxf(va, vb));
            float ea = __expf((va - m) * inv3);
            float eb = k9 ? __expf((vb - m) * inv3) : 0.f;
            float inv = 1.0f / rowsum16(ea + eb);
            int row = mt * 16 + (hi ? r + 8 : r);
            sAttn[wv][row * 32 + l15]      = ea * inv;
            sAttn[wv][row * 32 + 16 + l15] = eb * inv;  // cols 25..31 come out exactly 0
        }
    }
    asm volatile("" ::: "memory");

    // ============== out = attn * V + residual  (C pre-loaded with residuals) =========
    const float* AT = &sAttn[wv][0];
    const float* VT = &sValT[wv][0];
    v8f o0 = {0,0,0,0,0,0,0,0}, o1 = o0;
    #pragma unroll
    for (int r = 0; r < 8; ++r) {
        int rw = hi ? r + 8 : r;
        o0[r] = sRes[wv][rw * 16 + l15];
        o1[r] = sRes[wv][(16 + rw) * 16 + l15];
    }
    const int nn = k9 ? l15 : 0;        // clamp unread V columns (cols >= 9 discarded)
    #pragma unroll
    for (int kc = 0; kc < 7; ++kc) {    // contraction padded 25 -> 28 (attn pads are 0)
        int ss = kc * 4 + hi * 2;
        v2f bb = *(const v2f*)&VT[nn * 32 + ss];
        v2f a0 = *(const v2f*)&AT[l15 * 32 + ss];
        v2f a1 = *(const v2f*)&AT[(16 + l15) * 32 + ss];
        o0 = WMMA4(a0, bb, o0);
        o1 = WMMA4(a1, bb, o1);
    }

    // ================= LayerNorm over kv=9 and store ================================
    const float g  = P.p[25][k9 ? l15 : 0];
    const float be = P.p[26][k9 ? l15 : 0];
    #pragma unroll
    for (int mt = 0; mt < 2; ++mt) {
        v8f o = mt ? o1 : o0;
        #pragma unroll
        for (int r = 0; r < 8; ++r) {
            float v   = o[r];
            float mean = rowsum16(k9 ? v : 0.f) * (1.0f / 9.0f);
            float d   = k9 ? (v - mean) : 0.f;
            float var = rowsum16(d * d) * (1.0f / 9.0f);
            float y = (v - mean) * rsqrtf(var + LN_EPS) * g + be;
            int row = mt * 16 + (hi ? r + 8 : r);
            if (k9 && row < 25)
                out[((size_t)b * 25 + row) * 9 + l15] = y;
        }
    }
}

extern "C" void kernel_launch(void* const* d_in, const int* in_sizes, int n_in,
                              void* d_out, int out_size, void* d_ws, size_t ws_size,
                              hipStream_t stream)
{
    (void)d_ws; (void)ws_size; (void)out_size;
    Ptrs P;
    for (int i = 0; i < 27 && i < n_in; ++i) P.p[i] = (const float*)d_in[i];
    int nB = in_sizes[0] / 329;                 // 131072
    int blocks = (nB + WPB - 1) / WPB;
    attn_sub_module_kernel<<<blocks, NTHREADS, 0, stream>>>(P, (float*)d_out, nB);
}